// GNNLayer_35708358099443
// MI455X (gfx1250) — compile-verified
//
#include <hip/hip_runtime.h>

#define DD 128
#define NN 100000
#define LN_EPS 1e-5f

// K is 256 total -> 64 WMMA k-steps of 4; 8 n-tiles of 16 columns.
#define KSTEPS 64
#define NTILES 8

typedef __attribute__((ext_vector_type(2))) float v2f;
typedef __attribute__((ext_vector_type(8))) float v8f;

// ---------------------------------------------------------------------------
// Zero the x_neighbor accumulator (must happen every launch: ws is not
// re-poisoned between replays and spmm accumulates with atomics).
// ---------------------------------------------------------------------------
__global__ void gnn_zero_kernel(float4* __restrict__ p, int n4) {
    int i = blockIdx.x * blockDim.x + threadIdx.x;
    int stride = gridDim.x * blockDim.x;
    float4 z = {0.f, 0.f, 0.f, 0.f};
    for (; i < n4; i += stride) p[i] = z;
}

// ---------------------------------------------------------------------------
// SpMM scatter: one wave32 per edge. The wave streams the 128-float source
// row as coalesced float4 (512B burst, hot in 192MB L2 since x is 51MB) and
// scatters with hardware f32 atomics (global_atomic_add_f32).
// ---------------------------------------------------------------------------
__global__ void gnn_spmm_kernel(const float* __restrict__ x,
                                const int* __restrict__ erow,
                                const int* __restrict__ ecol,
                                const float* __restrict__ evals,
                                float* __restrict__ xn, int E) {
    int wave = (blockIdx.x * blockDim.x + threadIdx.x) >> 5;
    int lane = threadIdx.x & 31;
    if (wave >= E) return;
    int r = erow[wave];
    int c = ecol[wave];
    float a = evals[wave];
    const float4 xv =
        *reinterpret_cast<const float4*>(x + (size_t)c * DD + lane * 4);
    float* dst = xn + (size_t)r * DD + lane * 4;
    unsafeAtomicAdd(dst + 0, a * xv.x);
    unsafeAtomicAdd(dst + 1, a * xv.y);
    unsafeAtomicAdd(dst + 2, a * xv.z);
    unsafeAtomicAdd(dst + 3, a * xv.w);
}

// ---------------------------------------------------------------------------
// Repack W [256,128] row-major into per-lane WMMA B-fragment order:
//   Wp[((nt*KSTEPS + ks)*32 + lane)] = { W[k0+2*half][nt*16+lm],
//                                        W[k0+2*half+1][nt*16+lm] }  (float2)
// where k0 = 4*ks, half = lane>>4, lm = lane&15. One b64 load per k-step
// in the GEMM inner loop, fully coalesced (256B per wave per step).
// Total: 8*64*32 float2 = 128KB, L2-resident and reused by all 6250 blocks.
// ---------------------------------------------------------------------------
__global__ void gnn_packW_kernel(const float* __restrict__ W,
                                 float* __restrict__ Wp) {
    int t = blockIdx.x * blockDim.x + threadIdx.x;   // 0 .. 8*64*32-1
    if (t >= NTILES * KSTEPS * 32) return;
    int lane = t & 31;
    int ks   = (t >> 5) & (KSTEPS - 1);
    int nt   = t >> 11;
    int half = lane >> 4;
    int lm   = lane & 15;
    int k0   = 4 * ks;
    int col  = nt * 16 + lm;
    Wp[(size_t)t * 2 + 0] = W[(size_t)(k0 + 2 * half) * DD + col];
    Wp[(size_t)t * 2 + 1] = W[(size_t)(k0 + 2 * half + 1) * DD + col];
}

// ---------------------------------------------------------------------------
// Fused GEMM (WMMA f32 16x16x4) + bias + ReLU + residual + LayerNorm.
// Block = 256 threads (8 waves). Block owns 16 output rows; wave w owns the
// 16-column tile starting at 16*w. K = 256 split as 128 from x, 128 from xn.
//
// A fragment (16x4 f32): lane<16 holds row (m0+lane), K={k0,k0+1};
//                        lane>=16 holds row (m0+lane-16), K={k0+2,k0+3}.
// B fragment: preloaded in packed order (one v2f per lane per step).
// C/D (16x16 f32, 8 VGPRs): vgpr r, lanes 0-15 -> M=r, lanes 16-31 -> M=8+r.
// ---------------------------------------------------------------------------
__device__ __forceinline__ void gnn_epilogue_ln(
    float* h, const v8f& acc, const float* __restrict__ x,
    const float* __restrict__ bias, const float* __restrict__ gamma,
    const float* __restrict__ beta, float* __restrict__ out,
    int m0, int nb, int lm, int half) {
    // ---- epilogue: bias + ReLU + residual, stage to LDS ----
    const int col = nb + lm;
    const float bv = bias[col];
    #pragma unroll
    for (int r = 0; r < 8; ++r) {
        int mrow = r + 8 * half;
        float v = acc[r] + bv;
        v = v > 0.f ? v : 0.f;
        v += x[(size_t)(m0 + mrow) * DD + col];  // residual
        h[mrow * 132 + col] = v;
    }
    __syncthreads();

    // ---- LayerNorm over D=128: 16 threads per row, 8 cols each ----
    const int r   = threadIdx.x >> 4;   // 0..15
    const int seg = threadIdx.x & 15;   // 0..15
    const float* hr = h + r * 132 + seg * 8;
    float4 v0 = *reinterpret_cast<const float4*>(hr);
    float4 v1 = *reinterpret_cast<const float4*>(hr + 4);
    float s  = v0.x + v0.y + v0.z + v0.w + v1.x + v1.y + v1.z + v1.w;
    float ss = v0.x * v0.x + v0.y * v0.y + v0.z * v0.z + v0.w * v0.w +
               v1.x * v1.x + v1.y * v1.y + v1.z * v1.z + v1.w * v1.w;
    #pragma unroll
    for (int m = 8; m >= 1; m >>= 1) {
        s  += __shfl_xor(s,  m, 16);
        ss += __shfl_xor(ss, m, 16);
    }
    const float mu   = s * (1.0f / 128.0f);
    const float var  = ss * (1.0f / 128.0f) - mu * mu;
    const float rstd = rsqrtf(var + LN_EPS);

    const int cb = seg * 8;
    float4 g0 = *reinterpret_cast<const float4*>(gamma + cb);
    float4 g1 = *reinterpret_cast<const float4*>(gamma + cb + 4);
    float4 b0 = *reinterpret_cast<const float4*>(beta + cb);
    float4 b1 = *reinterpret_cast<const float4*>(beta + cb + 4);

    float4 o0, o1;
    o0.x = (v0.x - mu) * rstd * g0.x + b0.x;
    o0.y = (v0.y - mu) * rstd * g0.y + b0.y;
    o0.z = (v0.z - mu) * rstd * g0.z + b0.z;
    o0.w = (v0.w - mu) * rstd * g0.w + b0.w;
    o1.x = (v1.x - mu) * rstd * g1.x + b1.x;
    o1.y = (v1.y - mu) * rstd * g1.y + b1.y;
    o1.z = (v1.z - mu) * rstd * g1.z + b1.z;
    o1.w = (v1.w - mu) * rstd * g1.w + b1.w;

    float* orow = out + (size_t)(m0 + r) * DD + cb;
    *reinterpret_cast<float4*>(orow)     = o0;
    *reinterpret_cast<float4*>(orow + 4) = o1;
}

// Packed-B variant: one b64 A load + one b64 B load + one WMMA per k-step.
__global__ void __launch_bounds__(256)
gnn_gemm_ln_packed(const float* __restrict__ x,
                   const float* __restrict__ xn,
                   const float* __restrict__ Wp,     // packed fragments
                   const float* __restrict__ bias,
                   const float* __restrict__ gamma,
                   const float* __restrict__ beta,
                   float* __restrict__ out) {
    __shared__ float h[16 * 132];

    const int lane = threadIdx.x & 31;
    const int wv   = threadIdx.x >> 5;   // 0..7 -> n tile
    const int half = lane >> 4;
    const int lm   = lane & 15;
    const int m0   = blockIdx.x * 16;
    const int nb   = wv * 16;

    v8f acc = {};
    const v2f* __restrict__ Wf =
        reinterpret_cast<const v2f*>(Wp) + (size_t)wv * KSTEPS * 32 + lane;

    // ---- K = 0..127 : A from x, steps 0..31 ----
    const float* Arow = x + (size_t)(m0 + lm) * DD + 2 * half;
    #pragma unroll 8
    for (int s = 0; s < 32; ++s) {
        v2f a = *reinterpret_cast<const v2f*>(Arow + 4 * s);
        v2f b = Wf[s * 32];
        acc = __builtin_amdgcn_wmma_f32_16x16x4_f32(
            false, a, false, b, (short)0, acc, false, false);
    }
    // ---- K = 128..255 : A from x_neighbor, steps 32..63 ----
    const float* Nrow = xn + (size_t)(m0 + lm) * DD + 2 * half;
    #pragma unroll 8
    for (int s = 0; s < 32; ++s) {
        v2f a = *reinterpret_cast<const v2f*>(Nrow + 4 * s);
        v2f b = Wf[(s + 32) * 32];
        acc = __builtin_amdgcn_wmma_f32_16x16x4_f32(
            false, a, false, b, (short)0, acc, false, false);
    }

    gnn_epilogue_ln(h, acc, x, bias, gamma, beta, out, m0, nb, lm, half);
}

// Fallback: direct (unpacked) B loads, used if d_ws lacks room for Wp.
__global__ void __launch_bounds__(256)
gnn_gemm_ln_direct(const float* __restrict__ x,
                   const float* __restrict__ xn,
                   const float* __restrict__ W,
                   const float* __restrict__ bias,
                   const float* __restrict__ gamma,
                   const float* __restrict__ beta,
                   float* __restrict__ out) {
    __shared__ float h[16 * 132];

    const int lane = threadIdx.x & 31;
    const int wv   = threadIdx.x >> 5;
    const int half = lane >> 4;
    const int lm   = lane & 15;
    const int m0   = blockIdx.x * 16;
    const int nb   = wv * 16;

    v8f acc = {};
    const float* Arow = x + (size_t)(m0 + lm) * DD;
    #pragma unroll 8
    for (int k0 = 0; k0 < DD; k0 += 4) {
        v2f a = *reinterpret_cast<const v2f*>(Arow + k0 + 2 * half);
        v2f b;
        b[0] = W[(size_t)(k0 + 2 * half) * DD + nb + lm];
        b[1] = W[(size_t)(k0 + 2 * half + 1) * DD + nb + lm];
        acc = __builtin_amdgcn_wmma_f32_16x16x4_f32(
            false, a, false, b, (short)0, acc, false, false);
    }
    const float* Nrow = xn + (size_t)(m0 + lm) * DD;
    #pragma unroll 8
    for (int k0 = 0; k0 < DD; k0 += 4) {
        v2f a = *reinterpret_cast<const v2f*>(Nrow + k0 + 2 * half);
        v2f b;
        b[0] = W[(size_t)(DD + k0 + 2 * half) * DD + nb + lm];
        b[1] = W[(size_t)(DD + k0 + 2 * half + 1) * DD + nb + lm];
        acc = __builtin_amdgcn_wmma_f32_16x16x4_f32(
            false, a, false, b, (short)0, acc, false, false);
    }

    gnn_epilogue_ln(h, acc, x, bias, gamma, beta, out, m0, nb, lm, half);
}

// ---------------------------------------------------------------------------
extern "C" void kernel_launch(void* const* d_in, const int* in_sizes, int n_in,
                              void* d_out, int out_size, void* d_ws, size_t ws_size,
                              hipStream_t stream) {
    const float* x     = (const float*)d_in[0];
    const int*   erow  = (const int*)d_in[1];
    const int*   ecol  = (const int*)d_in[2];
    const float* evals = (const float*)d_in[3];
    const float* W     = (const float*)d_in[4];
    const float* bias  = (const float*)d_in[5];
    const float* gamma = (const float*)d_in[6];
    const float* beta  = (const float*)d_in[7];
    float* out = (float*)d_out;

    float* xn = (float*)d_ws;                          // [NN, DD], 51.2 MB
    const size_t xn_bytes = (size_t)NN * DD * sizeof(float);
    const size_t wp_bytes = (size_t)NTILES * KSTEPS * 32 * 2 * sizeof(float);

    const int E  = in_sizes[1];
    const int n4 = NN * DD / 4;

    gnn_zero_kernel<<<1024, 256, 0, stream>>>((float4*)xn, n4);

    const int wavesPerBlock = 8;  // 256 threads, wave32
    const int spmmBlocks = (E + wavesPerBlock - 1) / wavesPerBlock;
    gnn_spmm_kernel<<<spmmBlocks, 256, 0, stream>>>(x, erow, ecol, evals, xn, E);

    if (ws_size >= xn_bytes + wp_bytes) {
        float* Wp = (float*)((char*)d_ws + xn_bytes);
        const int packThreads = NTILES * KSTEPS * 32;   // 16384
        gnn_packW_kernel<<<packThreads / 256, 256, 0, stream>>>(W, Wp);
        gnn_gemm_ln_packed<<<NN / 16, 256, 0, stream>>>(x, xn, Wp, bias, gamma,
                                                        beta, out);
    } else {
        gnn_gemm_ln_direct<<<NN / 16, 256, 0, stream>>>(x, xn, W, bias, gamma,
                                                        beta, out);
    }
}